// Net_1786706395262
// MI455X (gfx1250) — compile-verified
//
#include <hip/hip_runtime.h>
#include <math.h>

typedef __attribute__((ext_vector_type(2))) float v2f;
typedef __attribute__((ext_vector_type(8))) float v8f;

#define FDIM 128

// ---------------------------------------------------------------------------
// K0: zero the feature accumulator (lives in d_out emb region) and init degree
//     to 1.0 (the self-loop weight), so the degree pass only handles edges.
// ---------------------------------------------------------------------------
__global__ void k_init(float* __restrict__ acc, float* __restrict__ deg,
                       long accN, int N) {
  long i = (long)blockIdx.x * blockDim.x + threadIdx.x;
  long stride = (long)gridDim.x * blockDim.x;
  for (long j = i; j < accN; j += stride) acc[j] = 0.0f;
  for (long j = i; j < N; j += stride) deg[j] = 1.0f;
}

// ---------------------------------------------------------------------------
// K1: degree on target nodes: deg[col[e]] += ew[e]
// ---------------------------------------------------------------------------
__global__ void k_degree(const int* __restrict__ cols,
                         const float* __restrict__ ew,
                         float* __restrict__ deg, int E) {
  int e = blockIdx.x * blockDim.x + threadIdx.x;
  if (e < E) atomicAdd(&deg[cols[e]], ew[e]);
}

// ---------------------------------------------------------------------------
// K2: deg -> dinv = rsqrt(deg) (deg >= 1 always, guard kept for parity)
// ---------------------------------------------------------------------------
__global__ void k_dinv(float* __restrict__ d, int N) {
  int i = blockIdx.x * blockDim.x + threadIdx.x;
  if (i < N) {
    float v = d[i];
    d[i] = (v > 0.0f) ? rsqrtf(v) : 0.0f;
  }
}

// ---------------------------------------------------------------------------
// K3: XW = X @ W1 via V_WMMA_F32_16X16X4_F32.
// Block = 256 threads (8 waves). Block computes a 16x128 output slab:
// X tile (16x128) staged in LDS once, each wave owns one 16x16 N-tile and
// iterates K in steps of 4 (32 WMMA issues per wave).
// A frag (16x4, 32b): lanes 0-15 hold K=k..k+1, lanes 16-31 hold K=k+2..k+3.
// B frag (4x16, 32b): mirrored layout. C/D: v8f, VGPR r -> row r (+8 hi half).
// ---------------------------------------------------------------------------
__global__ void __launch_bounds__(256)
k_gemm_wmma(const float* __restrict__ X, const float* __restrict__ W,
            float* __restrict__ XW, int N) {
  __shared__ float tileX[16 * FDIM];  // 8 KB

  const int m0 = blockIdx.x * 16;
  const int tid = threadIdx.x;

  // Stage 16x128 rows of X into LDS (clamp rows for the tail tile).
  for (int i = tid; i < 16 * FDIM / 4; i += 256) {
    int r = i >> 5;               // 32 float4 per row
    int c4 = i & 31;
    int row = m0 + r;
    if (row >= N) row = N - 1;
    const float4 v = *(const float4*)(X + (size_t)row * FDIM + c4 * 4);
    *(float4*)(tileX + r * FDIM + c4 * 4) = v;
  }
  __syncthreads();

  const int wave = tid >> 5;      // 0..7 -> which 16-col N tile
  const int lane = tid & 31;
  const int n0 = wave * 16;
  const int half = lane >> 4;     // 0: lanes 0-15, 1: lanes 16-31
  const int l16 = lane & 15;

  v8f c = {};
  for (int k = 0; k < FDIM; k += 4) {
    const int ka = k + half * 2;
    v2f a, b;
    a.x = tileX[l16 * FDIM + ka];
    a.y = tileX[l16 * FDIM + ka + 1];
    b.x = W[(size_t)ka * FDIM + n0 + l16];
    b.y = W[(size_t)(ka + 1) * FDIM + n0 + l16];
    c = __builtin_amdgcn_wmma_f32_16x16x4_f32(
        /*neg_a=*/false, a, /*neg_b=*/false, b,
        /*c_mod=*/(short)0, c, /*reuse_a=*/false, /*reuse_b=*/false);
  }

#pragma unroll
  for (int r = 0; r < 8; ++r) {
    int row = m0 + r + half * 8;
    if (row < N) XW[(size_t)row * FDIM + n0 + l16] = c[r];
  }
}

// ---------------------------------------------------------------------------
// K4: layer-1 message scatter. One wave per edge:
//   acc[col] += dinv[row]*ew*dinv[col] * XW[row]   (128 features)
// Lane l owns features 4l..4l+3 -> float4 gather + 4 f32 atomics.
// ---------------------------------------------------------------------------
__global__ void k_scatter1(const int* __restrict__ rows,
                           const int* __restrict__ cols,
                           const float* __restrict__ ew,
                           const float* __restrict__ dinv,
                           const float* __restrict__ xw,
                           float* __restrict__ acc, int E) {
  long gid = (long)blockIdx.x * blockDim.x + threadIdx.x;
  int e = (int)(gid >> 5);
  int lane = (int)(gid & 31);
  if (e >= E) return;
  const int r = rows[e];
  const int t = cols[e];
  const float nrm = dinv[r] * ew[e] * dinv[t];
  const float4 v = *(const float4*)(xw + (size_t)r * FDIM + lane * 4);
  float* dst = acc + (size_t)t * FDIM + lane * 4;
  atomicAdd(dst + 0, nrm * v.x);
  atomicAdd(dst + 1, nrm * v.y);
  atomicAdd(dst + 2, nrm * v.z);
  atomicAdd(dst + 3, nrm * v.w);
}

// ---------------------------------------------------------------------------
// K5: per-node epilogue of layer 1 + head of layer 2. One wave per node:
//   h = elu(acc + dinv^2 * xw_self + b1)   -> emb (in place over acc)
//   s[i] = h . W2                          (wave shuffle reduction)
//   y[i] = dinv[i]^2 * s[i]                (layer-2 self-loop seed)
// ---------------------------------------------------------------------------
__device__ __forceinline__ float elu1(float x) {
  return x > 0.0f ? x : (expf(x) - 1.0f);
}

__global__ void k_node(float* __restrict__ emb,        // acc in, h out (N*128)
                       const float* __restrict__ xw,
                       const float* __restrict__ dinv,
                       const float* __restrict__ b1,
                       const float* __restrict__ W2,
                       float* __restrict__ s,
                       float* __restrict__ y, int N) {
  long gid = (long)blockIdx.x * blockDim.x + threadIdx.x;
  int i = (int)(gid >> 5);
  int lane = (int)(gid & 31);
  if (i >= N) return;
  const float di = dinv[i];
  const float d2 = di * di;
  const size_t base = (size_t)i * FDIM + lane * 4;
  const float4 a = *(const float4*)(emb + base);
  const float4 xv = *(const float4*)(xw + base);
  const float4 bb = *(const float4*)(b1 + lane * 4);
  const float4 w2 = *(const float4*)(W2 + lane * 4);
  float4 h;
  h.x = elu1(a.x + d2 * xv.x + bb.x);
  h.y = elu1(a.y + d2 * xv.y + bb.y);
  h.z = elu1(a.z + d2 * xv.z + bb.z);
  h.w = elu1(a.w + d2 * xv.w + bb.w);
  *(float4*)(emb + base) = h;
  float partial = h.x * w2.x + h.y * w2.y + h.z * w2.z + h.w * w2.w;
#pragma unroll
  for (int off = 16; off > 0; off >>= 1)
    partial += __shfl_down(partial, off, 32);
  if (lane == 0) {
    s[i] = partial;
    y[i] = d2 * partial;  // self-loop contribution of layer 2
  }
}

// ---------------------------------------------------------------------------
// K6: layer-2 scalar scatter: y[col] += norm * s[row]. One thread per edge.
// ---------------------------------------------------------------------------
__global__ void k_scatter2(const int* __restrict__ rows,
                           const int* __restrict__ cols,
                           const float* __restrict__ ew,
                           const float* __restrict__ dinv,
                           const float* __restrict__ s,
                           float* __restrict__ y, int E) {
  int e = blockIdx.x * blockDim.x + threadIdx.x;
  if (e >= E) return;
  const int r = rows[e];
  const int t = cols[e];
  const float nrm = dinv[r] * ew[e] * dinv[t];
  atomicAdd(&y[t], nrm * s[r]);
}

// ---------------------------------------------------------------------------
// K7: out = sigmoid(y + b2) in place.
// ---------------------------------------------------------------------------
__global__ void k_out(float* __restrict__ y, const float* __restrict__ b2,
                      int N) {
  int i = blockIdx.x * blockDim.x + threadIdx.x;
  if (i < N) {
    float v = y[i] + b2[0];
    y[i] = 1.0f / (1.0f + expf(-v));
  }
}

// ---------------------------------------------------------------------------
extern "C" void kernel_launch(void* const* d_in, const int* in_sizes, int n_in,
                              void* d_out, int out_size, void* d_ws,
                              size_t ws_size, hipStream_t stream) {
  const float* x   = (const float*)d_in[0];
  const int*  eidx = (const int*)  d_in[1];
  const float* ew  = (const float*)d_in[2];
  const float* W1  = (const float*)d_in[3];
  const float* b1  = (const float*)d_in[4];
  const float* W2  = (const float*)d_in[5];
  const float* b2  = (const float*)d_in[6];

  const int N = in_sizes[0] / FDIM;
  const int E = in_sizes[2];
  const int* rows = eidx;       // edge_index[0]
  const int* cols = eidx + E;   // edge_index[1]

  float* out = (float*)d_out;   // [N]   layer-2 accumulator -> sigmoid output
  float* emb = out + N;         // [N*128] layer-1 accumulator -> elu(h)

  float* xw   = (float*)d_ws;               // [N*128]
  float* dinv = xw + (size_t)N * FDIM;      // [N] deg -> dinv in place
  float* s    = dinv + N;                   // [N]

  const int T = 256;

  // K0: acc = 0 (emb region), deg = 1 (self loops)
  k_init<<<2048, T, 0, stream>>>(emb, dinv, (long)N * FDIM, N);
  // K1: deg += ew on targets
  k_degree<<<(E + T - 1) / T, T, 0, stream>>>(cols, ew, dinv, E);
  // K2: dinv = rsqrt(deg)
  k_dinv<<<(N + T - 1) / T, T, 0, stream>>>(dinv, N);
  // K3: xw = x @ W1   (WMMA fp32)
  k_gemm_wmma<<<(N + 15) / 16, T, 0, stream>>>(x, W1, xw, N);
  // K4: feature scatter (one wave per edge)
  {
    long waves = (long)E;
    long blocks = (waves * 32 + T - 1) / T;
    k_scatter1<<<(int)blocks, T, 0, stream>>>(rows, cols, ew, dinv, xw, emb, E);
  }
  // K5: self-loop + bias + ELU + emb + s = h.W2, seed y
  {
    long blocks = ((long)N * 32 + T - 1) / T;
    k_node<<<(int)blocks, T, 0, stream>>>(emb, xw, dinv, b1, W2, s, out, N);
  }
  // K6: scalar scatter for layer 2
  k_scatter2<<<(E + T - 1) / T, T, 0, stream>>>(rows, cols, ew, dinv, s, out, E);
  // K7: sigmoid
  k_out<<<(N + T - 1) / T, T, 0, stream>>>(out, b2, N);
}